// SNNModel_78194174591402
// MI455X (gfx1250) — compile-verified
//
#include <hip/hip_runtime.h>
#include <hip/hip_bf16.h>

// ---------------------------------------------------------------------------
// SNN forward (Norse LIF/LI) for MI455X / gfx1250, fp32 WMMA path.
//
//   K_A : cin[t,b,h] = x[t,b,:] @ w_in[h,:]      (parallel, 32768x1024, K=700)
//         16x64 per wave, grouped loads, partial-wait pipelining.
//   K_S : per timestep (256 launches, prev/cur double buffer):
//         recurrent blocks: w_rec k-slab (KC=128) async-DMA'd into
//                           double-buffered LDS (shared by 8 waves);
//                           z read directly from L2 (no inter-wave reuse);
//                           i_cur = 0.8*i + cin[t] + z_prev @ w_rec^T,
//                           z_cur,v_cur = LIF(v_prev,i_prev)
//         readout block   : io_cur = 0.8*io + z_new(v,i) @ w_out^T,
//                           vo_cur = 0.9*vo + 0.1*io
//   K_L : log_softmax(vo_final) -> d_out [128,20]
// ---------------------------------------------------------------------------

typedef __attribute__((ext_vector_type(2))) float v2f;
typedef __attribute__((ext_vector_type(8))) float v8f;

#define T_STEPS 256
#define BATCH   128
#define I_DIM   700
#define H_DIM   1024
#define O_DIM   20

#define KC      128             // k-chunk staged in LDS (8 chunks -> 8 barriers/step)
#define WSTR    132             // LDS row stride in floats (528B: 16B-aligned rows,
                                // 132 mod 64 = 4 -> lanes 0..15 hit distinct banks)

__device__ __forceinline__ float2 ldg2(const float* p) {
    return *reinterpret_cast<const float2*>(p);
}

__device__ __forceinline__ v2f lds2(const float* p) {
    return *reinterpret_cast<const v2f*>(p);
}

__device__ __forceinline__ v8f wmma_f32(v2f a, v2f b, v8f c) {
    return __builtin_amdgcn_wmma_f32_16x16x4_f32(false, a, false, b,
                                                 (short)0, c, false, false);
}

// Async DMA of 16 bytes/lane from global to LDS (ASYNCcnt-tracked).
__device__ __forceinline__ void async_copy16(const float* lds_dst,
                                             const float* gsrc) {
    asm volatile("global_load_async_to_lds_b128 %0, %1, off"
                 :: "v"((unsigned int)(unsigned long long)lds_dst),
                    "v"((unsigned long long)gsrc)
                 : "memory");
}

__device__ __forceinline__ void wait_async0() {
    asm volatile("s_wait_asynccnt 0x0" ::: "memory");
}

// ---------------------------------------------------------------------------
// Kernel A: cin = x @ w_in^T.  grid = (256, 16), block = 256 (8 waves).
// Wave computes 16(M) x 64(N): 1 A-frag reused across 4 B-frags; 2 k-steps per
// iteration with all loads grouped ahead of the 8 WMMAs.
// ---------------------------------------------------------------------------
__global__ void snn_in_gemm(const float* __restrict__ x,
                            const float* __restrict__ w_in,
                            float* __restrict__ cin) {
    const int wave = threadIdx.x >> 5;
    const int lane = threadIdx.x & 31;
    const int half = lane >> 4;
    const int ml   = lane & 15;
    const int kb   = 2 * half;

    const int m0 = (blockIdx.x * 8 + wave) * 16;   // M tile
    const int n0 = blockIdx.y * 64;                // N range (4 tiles of 16)

    const float* arow = x + (size_t)(m0 + ml) * I_DIM;
    const float* bcol[4];
#pragma unroll
    for (int j = 0; j < 4; ++j)
        bcol[j] = w_in + (size_t)(n0 + 16 * j + ml) * I_DIM;

    v8f acc[4] = {};

    // main loop: 87 iterations x 8 floats (2 k-steps), loads grouped.
    // unroll(1): keep single exit -> no accumulator live-range copies.
    int k = 0;
#pragma unroll 1
    for (; k + 8 <= I_DIM; k += 8) {
        v2f a0, a1, b0[4], b1[4];
        {
            float2 t0 = ldg2(arow + k + kb);
            float2 t1 = ldg2(arow + k + 4 + kb);
            a0.x = t0.x; a0.y = t0.y; a1.x = t1.x; a1.y = t1.y;
        }
#pragma unroll
        for (int j = 0; j < 4; ++j) {
            float2 t0 = ldg2(bcol[j] + k + kb);
            float2 t1 = ldg2(bcol[j] + k + 4 + kb);
            b0[j].x = t0.x; b0[j].y = t0.y; b1[j].x = t1.x; b1[j].y = t1.y;
        }
#pragma unroll
        for (int j = 0; j < 4; ++j) acc[j] = wmma_f32(a0, b0[j], acc[j]);
#pragma unroll
        for (int j = 0; j < 4; ++j) acc[j] = wmma_f32(a1, b1[j], acc[j]);
    }
    // tail (700 = 87*8 + 4): one k-step
    {
        float2 ta = ldg2(arow + k + kb);
        v2f a; a.x = ta.x; a.y = ta.y;
#pragma unroll
        for (int j = 0; j < 4; ++j) {
            float2 tb = ldg2(bcol[j] + k + kb);
            v2f b; b.x = tb.x; b.y = tb.y;
            acc[j] = wmma_f32(a, b, acc[j]);
        }
    }

    // D layout: VGPR r -> M = r + 8*half, N = lane&15
    float* out = cin + (size_t)(m0 + 8 * half) * H_DIM + n0 + ml;
#pragma unroll
    for (int j = 0; j < 4; ++j)
#pragma unroll
        for (int r = 0; r < 8; ++r)
            out[(size_t)r * H_DIM + 16 * j] = acc[j][r];
}

// ---------------------------------------------------------------------------
// Kernel S: one timestep.  grid = 33 blocks x 256 threads.
//   blocks 0..31 : recurrent; block owns M=128 x N=32 (waves = M tiles).
//                  w_rec KC=128 slab async-DMA'd into double-buffered LDS;
//                  z loaded directly from global (L2-resident, no reuse).
//   block  32    : readout (z_new recomputed from v,i) + vo update.
// ---------------------------------------------------------------------------
__global__ void snn_step(const float* __restrict__ w_rec,
                         const float* __restrict__ w_out,
                         const float* __restrict__ cin_t,
                         const float* __restrict__ zp, const float* __restrict__ vp,
                         const float* __restrict__ ip, const float* __restrict__ vop,
                         const float* __restrict__ iop,
                         float* __restrict__ zc, float* __restrict__ vc,
                         float* __restrict__ ic, float* __restrict__ voc,
                         float* __restrict__ ioc) {
    __shared__ float wbuf[2][32 * WSTR];   // 2 x 16.5 KB

    const int tid  = threadIdx.x;
    const int wave = tid >> 5;
    const int lane = tid & 31;
    const int half = lane >> 4;
    const int ml   = lane & 15;
    const int kb   = 2 * half;

    if (blockIdx.x < 32) {
        // ------------------- recurrent part -------------------
        const int n0 = blockIdx.x * 32;   // 32 h-columns for this block
        const int m0 = wave * 16;         // 16 batch rows for this wave

        // LIF elementwise for this wave's (16 x 32) patch
        const size_t ebase = (size_t)(m0 + 8 * half) * H_DIM + n0 + ml;
#pragma unroll
        for (int r = 0; r < 8; ++r) {
#pragma unroll
            for (int j = 0; j < 2; ++j) {
                const size_t idx = ebase + (size_t)r * H_DIM + 16 * j;
                const float v_ = vp[idx], i_ = ip[idx];
                const float vdec = 0.9f * v_ + 0.1f * i_;
                const float zn = (vdec > 1.0f) ? 1.0f : 0.0f;
                vc[idx] = (1.0f - zn) * vdec;
                zc[idx] = zn;
            }
        }

        // acc init: 0.8*i_prev + cin[t]
        v8f acc0, acc1;
#pragma unroll
        for (int r = 0; r < 8; ++r) {
            const size_t idx = ebase + (size_t)r * H_DIM;
            acc0[r] = 0.8f * ip[idx] + cin_t[idx];
            acc1[r] = 0.8f * ip[idx + 16] + cin_t[idx + 16];
        }

        // staging layout: w slab 32 rows x 512B = 1024 x 16B, 4 copies/thread
        const float* gsrc[4];
        float*       dst0[4];
        float*       dst1[4];
#pragma unroll
        for (int j = 0; j < 4; ++j) {
            const int lin = j * 256 + tid;
            const int row = lin >> 5, c16 = lin & 31;
            gsrc[j] = w_rec + (size_t)(n0 + row) * H_DIM + c16 * 4;
            dst0[j] = &wbuf[0][row * WSTR + c16 * 4];
            dst1[j] = &wbuf[1][row * WSTR + c16 * 4];
        }

        // stage chunk 0
#pragma unroll
        for (int j = 0; j < 4; ++j) async_copy16(dst0[j], gsrc[j]);

        const float* zrow = zp + (size_t)(m0 + ml) * H_DIM;
        const int NCH = H_DIM / KC;   // 8 chunks
#pragma unroll 1
        for (int c = 0; c < NCH; ++c) {
            wait_async0();
            __syncthreads();

            if (c + 1 < NCH) {        // stage next chunk into other buffer
                const int k0 = (c + 1) * KC;
                float* const* dst = ((c + 1) & 1) ? dst1 : dst0;
#pragma unroll
                for (int j = 0; j < 4; ++j) async_copy16(dst[j], gsrc[j] + k0);
            }

            // compute from buffer c&1: 32 k-steps in groups of 4
            const float* wb = wbuf[c & 1];
            const float* zk = zrow + c * KC;
#pragma unroll
            for (int kc = 0; kc < KC; kc += 16) {
                v2f a[4];
#pragma unroll
                for (int u = 0; u < 4; ++u) {
                    float2 t = ldg2(zk + kc + 4 * u + kb);
                    a[u].x = t.x; a[u].y = t.y;
                }
#pragma unroll
                for (int u = 0; u < 4; ++u) {
                    v2f b0 = lds2(wb + ml * WSTR + kc + 4 * u + kb);
                    v2f b1 = lds2(wb + (16 + ml) * WSTR + kc + 4 * u + kb);
                    acc0 = wmma_f32(a[u], b0, acc0);
                    acc1 = wmma_f32(a[u], b1, acc1);
                }
            }
        }

#pragma unroll
        for (int r = 0; r < 8; ++r) {
            const size_t idx = ebase + (size_t)r * H_DIM;
            ic[idx]      = acc0[r];
            ic[idx + 16] = acc1[r];
        }
    } else {
        // ------------------- readout part -------------------
        const int b0 = wave * 16;          // 8 waves cover B=128
        const bool ov1 = (ml < 4);         // o tile 1: cols 16..19 valid

        v8f acc0, acc1;
#pragma unroll
        for (int r = 0; r < 8; ++r) {
            const int b = b0 + r + 8 * half;
            acc0[r] = 0.8f * iop[(size_t)b * O_DIM + ml];
            acc1[r] = ov1 ? 0.8f * iop[(size_t)b * O_DIM + 16 + ml] : 0.0f;
        }

        const float* vrow = vp + (size_t)(b0 + ml) * H_DIM;
        const float* irow = ip + (size_t)(b0 + ml) * H_DIM;
        const float* w0   = w_out + (size_t)ml * H_DIM;
        const float* w1   = w_out + (size_t)(ov1 ? 16 + ml : 0) * H_DIM;

#pragma unroll 1
        for (int k = 0; k < H_DIM; k += 16) {   // 4 k-steps, loads grouped
            v2f a[4], f0[4], f1[4];
#pragma unroll
            for (int u = 0; u < 4; ++u) {
                const int kk = k + 4 * u + kb;
                float2 vv = ldg2(vrow + kk);
                float2 ii = ldg2(irow + kk);
                a[u].x = (0.9f * vv.x + 0.1f * ii.x > 1.0f) ? 1.0f : 0.0f;
                a[u].y = (0.9f * vv.y + 0.1f * ii.y > 1.0f) ? 1.0f : 0.0f;
                float2 t0 = ldg2(w0 + kk);
                float2 t1 = ldg2(w1 + kk);
                f0[u].x = t0.x; f0[u].y = t0.y;
                f1[u].x = ov1 ? t1.x : 0.0f;
                f1[u].y = ov1 ? t1.y : 0.0f;
            }
#pragma unroll
            for (int u = 0; u < 4; ++u) {
                acc0 = wmma_f32(a[u], f0[u], acc0);
                acc1 = wmma_f32(a[u], f1[u], acc1);
            }
        }

#pragma unroll
        for (int r = 0; r < 8; ++r) {
            const int b = b0 + r + 8 * half;
            ioc[(size_t)b * O_DIM + ml] = acc0[r];
            if (ov1) ioc[(size_t)b * O_DIM + 16 + ml] = acc1[r];
        }

        // LI membrane update: 16 rows x 20 cols per wave
        for (int e = lane; e < 16 * O_DIM; e += 32) {
            const int b = b0 + e / O_DIM;
            const int o = e % O_DIM;
            const size_t idx = (size_t)b * O_DIM + o;
            voc[idx] = 0.9f * vop[idx] + 0.1f * iop[idx];
        }
    }
}

// ---------------------------------------------------------------------------
// Kernel L: log_softmax over final vo [128,20].  1 block x 128 threads.
// ---------------------------------------------------------------------------
__global__ void snn_logsoftmax(const float* __restrict__ vo,
                               float* __restrict__ out) {
    const int b = threadIdx.x;
    const float* row = vo + (size_t)b * O_DIM;
    float m = row[0];
    for (int o = 1; o < O_DIM; ++o) m = fmaxf(m, row[o]);
    float s = 0.0f;
    for (int o = 0; o < O_DIM; ++o) s += __expf(row[o] - m);
    const float lse = m + __logf(s);
    float* orow = out + (size_t)b * O_DIM;
    for (int o = 0; o < O_DIM; ++o) orow[o] = row[o] - lse;
}

// ---------------------------------------------------------------------------
extern "C" void kernel_launch(void* const* d_in, const int* in_sizes, int n_in,
                              void* d_out, int out_size, void* d_ws, size_t ws_size,
                              hipStream_t stream) {
    const float* x     = (const float*)d_in[0];
    const float* w_in  = (const float*)d_in[1];
    const float* w_rec = (const float*)d_in[2];
    const float* w_out = (const float*)d_in[3];

    float* ws = (float*)d_ws;

    const size_t TBH = (size_t)T_STEPS * BATCH * H_DIM;   // cin
    const size_t BH  = (size_t)BATCH * H_DIM;
    const size_t BO  = (size_t)BATCH * O_DIM;
    const size_t SB  = 3 * BH + 2 * BO;                   // one state block

    float* cin = ws;
    float* st[2] = { ws + TBH, ws + TBH + SB };

    // zero initial state (block 0: z0,v0,i0,vo0,io0)
    hipMemsetAsync(st[0], 0, SB * sizeof(float), stream);

    // Phase 1: hoisted input GEMM (fully parallel)
    snn_in_gemm<<<dim3((T_STEPS * BATCH) / 128, H_DIM / 64), 256, 0, stream>>>(
        x, w_in, cin);

    // Phase 2: sequential scan, one kernel per step, ping-pong state
    for (int t = 0; t < T_STEPS; ++t) {
        float* sp = st[t & 1];
        float* sc = st[(t + 1) & 1];
        snn_step<<<33, 256, 0, stream>>>(
            w_rec, w_out, cin + (size_t)t * BH,
            /*zp*/ sp, /*vp*/ sp + BH, /*ip*/ sp + 2 * BH,
            /*vop*/ sp + 3 * BH, /*iop*/ sp + 3 * BH + BO,
            /*zc*/ sc, /*vc*/ sc + BH, /*ic*/ sc + 2 * BH,
            /*voc*/ sc + 3 * BH, /*ioc*/ sc + 3 * BH + BO);
    }

    // Phase 3: final vo lives in state block (T % 2) == 0
    snn_logsoftmax<<<1, BATCH, 0, stream>>>(st[0] + 3 * BH, (float*)d_out);
}